// HierarchicalDocumentMemory_70205535420748
// MI455X (gfx1250) — compile-verified
//
#include <hip/hip_runtime.h>
#include <hip/hip_bf16.h>

typedef float v2f __attribute__((ext_vector_type(2)));
typedef float v8f __attribute__((ext_vector_type(8)));

#define T_SENT 1000
#define T_PARA 200
#define DK 256
#define DV 256
#define DEMB 768
#define S_DECAY 0.95f
#define P_DECAY 0.99f
#define ALPHA 0.1f
#define EPS 1e-12f

// ---------------------------------------------------------------------------
// Kernel 1: K = emb@W_k + b_k ; V = emb@W_v + b_v  via V_WMMA_F32_16X16X4_F32.
// One wave per 16x16 output tile. 63 row tiles (1008 rows; overflow rows are
// computed on CLAMPED addresses and simply never stored -> no divergence in
// the inner loop). 32 col tiles (0..15 -> K projection, 16..31 -> V).
// A-frag (16x4 f32): lanes 0-15 hold M=0..15 / K={0,1}; lanes 16-31 K={2,3}.
// B-frag (4x16 f32): mirrored striping, N = lane&15.
// ---------------------------------------------------------------------------
__global__ void hdm_kv_proj_wmma(const float* __restrict__ emb,
                                 const float* __restrict__ Wk, const float* __restrict__ bk,
                                 const float* __restrict__ Wv, const float* __restrict__ bv,
                                 float* __restrict__ K, float* __restrict__ V) {
  const int wid  = (blockIdx.x * blockDim.x + threadIdx.x) >> 5;
  const int lane = threadIdx.x & 31;
  const int mT  = wid % 63;
  const int sel = wid / 63;              // wave-uniform
  const bool isV = sel >= 16;
  const int nT = sel & 15;
  const float* __restrict__ W = isV ? Wv : Wk;
  const float* __restrict__ b = isV ? bv : bk;
  float* __restrict__ O = isV ? V : K;

  const int half = lane >> 4;            // 0: lanes 0-15, 1: lanes 16-31
  const int l15  = lane & 15;
  const int m = mT * 16 + l15;           // A-fragment row (logical)
  const int mc = m < T_SENT ? m : (T_SENT - 1);   // clamped: branch-free loads
  const int n = nT * 16 + l15;           // B-fragment col

  const float* __restrict__ Arow = emb + (size_t)mc * DEMB;

  v8f acc = {};
  for (int kk = 0; kk < DEMB; kk += 4) {
    const int k0 = kk + half * 2;
    v2f a = *(const v2f*)(Arow + k0);    // 8B aligned: k0 even, row stride 768
    v2f bf;
    bf.x = W[(k0    ) * DK + n];
    bf.y = W[(k0 + 1) * DK + n];
    acc = __builtin_amdgcn_wmma_f32_16x16x4_f32(false, a, false, bf,
                                                (short)0, acc, false, false);
  }
  const float bias = b[n];
  #pragma unroll
  for (int j = 0; j < 8; ++j) {
    const int row = mT * 16 + half * 8 + j;   // C/D layout: VGPR j -> M=j (+8 hi lanes)
    if (row < T_SENT) O[row * DK + nT * 16 + l15] = acc[j] + bias;
  }
}

// ---------------------------------------------------------------------------
// Kernel 2: paragraph K/V. Projection is affine, so PK[p] = mean of 5 K rows.
// ---------------------------------------------------------------------------
__global__ void hdm_para_means(const float* __restrict__ K, const float* __restrict__ V,
                               float* __restrict__ PK, float* __restrict__ PV) {
  const int idx = blockIdx.x * 256 + threadIdx.x;   // exactly 200*256 threads
  const int p = idx >> 8, c = idx & 255;
  float sk = 0.f, sv = 0.f;
  #pragma unroll
  for (int j = 0; j < 5; ++j) {
    sk += K[(p * 5 + j) * DK + c];
    sv += V[(p * 5 + j) * DV + c];
  }
  PK[idx] = sk * 0.2f;
  PV[idx] = sv * 0.2f;
}

// ---------------------------------------------------------------------------
// Kernel 3: precompute 1/max(||k||, eps) for all 1200 keys (removes the
// per-step norm reduction from the serial scan).
// ---------------------------------------------------------------------------
__global__ void hdm_key_norms(const float* __restrict__ K, const float* __restrict__ PK,
                              float* __restrict__ rnS, float* __restrict__ rnP) {
  const int t = blockIdx.x * 256 + threadIdx.x;
  if (t < T_SENT) {
    const float* r = K + t * DK;
    float s = 0.f;
    for (int i = 0; i < DK; ++i) s = fmaf(r[i], r[i], s);
    rnS[t] = 1.f / fmaxf(sqrtf(s), EPS);
  } else if (t < T_SENT + T_PARA) {
    const float* r = PK + (t - T_SENT) * DK;
    float s = 0.f;
    for (int i = 0; i < DK; ++i) s = fmaf(r[i], r[i], s);
    rnP[t - T_SENT] = 1.f / fmaxf(sqrtf(s), EPS);
  }
}

// ---------------------------------------------------------------------------
// Kernel 4: the sequential delta-rule scans + doc level + combine.
// One block, 1024 threads. Thread (kg = tid>>8, v = tid&255) owns
// M[kg*64 + i][v] for i in 0..63 in REGISTERS. LDS only holds the broadcast
// key, 4x256 reduction partials, and row sums.
// Per step: M <- decay*M - alpha*kn*(kn^T (decay*M)) + kn v^T
//         = decay*M + kn * (v - alpha*decay*(kn^T M))
// Next step's K/V rows are prefetched (global_prefetch_b8) to hide the
// dependent-load latency at the top of each serial step.
// ---------------------------------------------------------------------------
__global__ void __launch_bounds__(1024)
hdm_scan(const float* __restrict__ K, const float* __restrict__ V,
         const float* __restrict__ PK, const float* __restrict__ PV,
         const float* __restrict__ rnS, const float* __restrict__ rnP,
         const float* __restrict__ q, float* __restrict__ out) {
  __shared__ float kbuf[256];
  __shared__ float sPart[1024];
  __shared__ float rows[256];
  __shared__ float misc[2];

  const int tid = threadIdx.x;
  const int v  = tid & 255;
  const int kg = tid >> 8;

  float m[64];
  float rec[2];

  for (int lvl = 0; lvl < 2; ++lvl) {
    const float* Kp = lvl ? PK : K;
    const float* Vp = lvl ? PV : V;
    const float* rn = lvl ? rnP : rnS;
    const int   T     = lvl ? T_PARA : T_SENT;
    const float decay = lvl ? P_DECAY : S_DECAY;

    #pragma unroll
    for (int i = 0; i < 64; ++i) m[i] = 0.f;

    for (int t = 0; t < T; ++t) {
      if (tid < 256) {
        kbuf[tid] = Kp[t * DK + tid] * rn[t];            // kn broadcast
        if (t + 1 < T) {
          __builtin_prefetch(&Kp[(t + 1) * DK + tid], 0, 0);  // global_prefetch_b8
          __builtin_prefetch(&Vp[(t + 1) * DV + tid], 0, 0);
        }
      }
      __syncthreads();
      float part = 0.f;
      #pragma unroll
      for (int i = 0; i < 64; ++i) part = fmaf(kbuf[kg * 64 + i], m[i], part);
      sPart[tid] = part;
      __syncthreads();
      const float s  = sPart[v] + sPart[256 + v] + sPart[512 + v] + sPart[768 + v];
      const float vv = Vp[t * DV + v];
      const float c1 = vv - ALPHA * decay * s;
      #pragma unroll
      for (int i = 0; i < 64; ++i) m[i] = fmaf(decay, m[i], kbuf[kg * 64 + i] * c1);
      __syncthreads();   // protect kbuf/sPart reuse next step
    }

    // recall(M) = qn^T M  (heads are identical -> mean over heads is a no-op)
    if (tid < 256) kbuf[tid] = q[tid];
    __syncthreads();
    if (tid == 0) {
      float sq = 0.f;
      for (int i = 0; i < 256; ++i) sq = fmaf(kbuf[i], kbuf[i], sq);
      misc[0] = 1.f / fmaxf(sqrtf(sq), EPS);
    }
    __syncthreads();
    const float qs = misc[0];
    float part = 0.f;
    #pragma unroll
    for (int i = 0; i < 64; ++i) part = fmaf(kbuf[kg * 64 + i] * qs, m[i], part);
    sPart[tid] = part;
    __syncthreads();
    rec[lvl] = sPart[v] + sPart[256 + v] + sPart[512 + v] + sPart[768 + v];
    __syncthreads();   // protect sPart before reuse
  }

  // ---- Document level (M_doc0 == 0): M_doc = kn_doc (x) doc_v -------------
  // doc_v[v] = column mean of M_para; doc_k[k] = row mean of M_para.
  float part2 = 0.f;
  #pragma unroll
  for (int i = 0; i < 64; ++i) part2 += m[i];
  sPart[tid] = part2;
  if (tid < 256) rows[tid] = 0.f;
  __syncthreads();
  const float colsum = sPart[v] + sPart[256 + v] + sPart[512 + v] + sPart[768 + v];
  const float docv = colsum * (1.f / 256.f);
  #pragma unroll
  for (int i = 0; i < 64; ++i) atomicAdd(&rows[kg * 64 + i], m[i]);   // ds_add_f32
  __syncthreads();
  if (tid == 0) {
    // kbuf still holds q, misc[0] = 1/max(||q||,eps)
    float sq = 0.f;
    for (int k = 0; k < 256; ++k) {
      const float dk = rows[k] * (1.f / 256.f);
      sq = fmaf(dk, dk, sq);
    }
    const float rdk = 1.f / fmaxf(sqrtf(sq), EPS);
    float dot = 0.f;
    for (int k = 0; k < 256; ++k)
      dot = fmaf(kbuf[k] * misc[0], rows[k] * (1.f / 256.f) * rdk, dot);
    misc[1] = dot;   // qn . kn_doc
  }
  __syncthreads();
  if (tid < 256)
    out[tid] = 0.2f * rec[0] + 0.3f * rec[1] + 0.5f * misc[1] * docv;
}

// ---------------------------------------------------------------------------
extern "C" void kernel_launch(void* const* d_in, const int* in_sizes, int n_in,
                              void* d_out, int out_size, void* d_ws, size_t ws_size,
                              hipStream_t stream) {
  const float* emb = (const float*)d_in[0];
  const float* q   = (const float*)d_in[1];
  const float* Wk  = (const float*)d_in[2];
  const float* bk  = (const float*)d_in[3];
  const float* Wv  = (const float*)d_in[4];
  const float* bv  = (const float*)d_in[5];
  // d_in[6..8] are the zero-initialized M0 tensors (zeros by construction).
  float* ws  = (float*)d_ws;
  float* K   = ws;                 // 1000*256
  float* V   = ws + 256000;        // 1000*256
  float* PK  = ws + 512000;        // 200*256
  float* PV  = ws + 563200;        // 200*256
  float* rnS = ws + 614400;        // 1000
  float* rnP = ws + 615400;        // 200
  float* out = (float*)d_out;      // 256

  // 63 row-tiles * 32 col-tiles = 2016 waves; 8 waves per 256-thread block.
  hdm_kv_proj_wmma<<<252, 256, 0, stream>>>(emb, Wk, bk, Wv, bv, K, V);
  hdm_para_means<<<200, 256, 0, stream>>>(K, V, PK, PV);
  hdm_key_norms<<<5, 256, 0, stream>>>(K, PK, rnS, rnP);
  hdm_scan<<<1, 1024, 0, stream>>>(K, V, PK, PV, rnS, rnP, q, out);
}